// MoNet_38448547234483
// MI455X (gfx1250) — compile-verified
//
#include <hip/hip_runtime.h>
#include <hip/hip_bf16.h>
#include <math.h>
#include <stdint.h>

#define V_N 163842
#define E_N 983040
#define COUT 32
#define KM   96    // K * COUT
#define HLD  128   // h row stride: 96 message cols + 32 root cols

typedef float v2f __attribute__((ext_vector_type(2)));
typedef float v8f __attribute__((ext_vector_type(8)));
typedef unsigned int u32x4 __attribute__((ext_vector_type(4)));
typedef int i32x4 __attribute__((ext_vector_type(4)));
typedef int i32x8 __attribute__((ext_vector_type(8)));

#if defined(__has_builtin)
#  if __has_builtin(__builtin_amdgcn_tensor_load_to_lds) && \
      __has_builtin(__builtin_amdgcn_s_wait_tensorcnt)
#    define HAVE_TDM 1
#  endif
#endif
#ifndef HAVE_TDM
#  define HAVE_TDM 0
#endif

// ---------------- utility kernels ----------------

__global__ void zero_f32(float* __restrict__ p, int n) {
  int i = blockIdx.x * blockDim.x + threadIdx.x;
  if (i < n) p[i] = 0.0f;
}

__global__ void deg_count(const int* __restrict__ dst, float* __restrict__ deg) {
  int e = blockIdx.x * blockDim.x + threadIdx.x;
  if (e < E_N) atomicAdd(&deg[dst[e]], 1.0f);
}

__global__ void deg_invert(float* __restrict__ deg) {
  int v = blockIdx.x * blockDim.x + threadIdx.x;
  if (v < V_N) deg[v] = 1.0f / fmaxf(deg[v], 1.0f);
}

// ---------------- fused GEMM: h[V,128] = X[V,cin] @ [g | root] ----------------
// cols 0..95   <- g[cin,96]    (message transform, K*COUT)
// cols 96..127 <- root[cin,32] (root transform)
// One block = 16 rows of X; 8 waves, wave w computes the 16x16 N-tile w.
// A tile arrives via the Tensor Data Mover (TDM) when available; the fused
// weight tile is staged to LDS once per block so the WMMA K-loop is
// branch-free (ds_load + v_wmma only).
__global__ __launch_bounds__(256) void gemm_wmma(
    const float* __restrict__ X, int ldx, int cin,
    const float* __restrict__ g, const float* __restrict__ root,
    float* __restrict__ h) {
  __shared__ float xs[16 * 32];     // A tile, K padded to 32 with zeros
  __shared__ float wsm[32 * 128];   // fused [g | root] tile, zero-padded K
  const int mbase = blockIdx.x * 16;
  const int tid = threadIdx.x;

  // stage fused weight tile (guards live here, outside the hot loop)
  for (int idx = tid; idx < 32 * 128; idx += 256) {
    int k = idx >> 7, c = idx & 127;
    float w = 0.0f;
    if (k < cin) w = (c < KM) ? g[k * KM + c] : root[k * COUT + (c - KM)];
    wsm[idx] = w;
  }

#if HAVE_TDM
  // A tile via TDM: 2D tile [tile_dim1=16 rows x tile_dim0=32 cols] from the
  // [V x cin] tensor (row stride ldx). OOB reads return zero, which provides
  // both the K-padding (cols cin..31) and the M-edge padding (rows >= V).
  if (tid == 0) {
    unsigned long long gaddr =
        (unsigned long long)(uintptr_t)X + (unsigned long long)mbase * (unsigned)ldx * 4ull;
    unsigned ldsa = (unsigned)(uintptr_t)&xs[0];
    unsigned dim0 = (unsigned)cin;            // tensor row length (elements)
    unsigned dim1 = (unsigned)(V_N - mbase);  // remaining rows
    u32x4 d0 = { 1u,                                              // count=1, load, user
                 ldsa,                                            // lds_addr
                 (unsigned)(gaddr & 0xffffffffu),                 // global_addr[31:0]
                 (unsigned)((gaddr >> 32) & 0x01ffffffu) | (2u << 30) };  // addr[56:32], type=2
    i32x8 d1 = { (int)0x00020000u,                         // data_size=2 (4B), no flags
                 (int)((dim0 & 0xffffu) << 16),            // tensor_dim0[15:0]
                 (int)((dim0 >> 16) | ((dim1 & 0xffffu) << 16)),  // dim0[31:16] | dim1[15:0]
                 (int)((dim1 >> 16) | (32u << 16)),        // dim1[31:16] | tile_dim0=32
                 (int)16,                                  // tile_dim1=16, tile_dim2=0
                 (int)ldx,                                 // tensor_dim0_stride[31:0]
                 0, 0 };                                   // stride hi, dim1_stride=0
    i32x4 dz = { 0, 0, 0, 0 };
#if __clang_major__ >= 23
    i32x8 dz8 = { 0, 0, 0, 0, 0, 0, 0, 0 };
    __builtin_amdgcn_tensor_load_to_lds(d0, d1, dz, dz, dz8, 0);
#else
    __builtin_amdgcn_tensor_load_to_lds(d0, d1, dz, dz, 0);
#endif
    __builtin_amdgcn_s_wait_tensorcnt(0);
  }
#else
  for (int idx = tid; idx < 16 * 32; idx += 256) {
    int r = idx >> 5, k = idx & 31;
    int row = mbase + r;
    xs[idx] = (row < V_N && k < cin) ? X[(size_t)row * ldx + k] : 0.0f;
  }
#endif
  __syncthreads();

  const int lane  = tid & 31;
  const int wid   = tid >> 5;                // n-tile index 0..7
  const int col   = wid * 16 + (lane & 15);  // output column 0..127
  const int khalf = (lane >> 4) * 2;         // lanes 0-15: K pair {0,1}; 16-31: {2,3}
  const int ksteps = (cin + 3) >> 2;

  const float* xrow  = &xs[(lane & 15) * 32 + khalf];
  const float* wcolk = &wsm[col + khalf * 128];

  v8f acc = {};
  for (int s = 0; s < ksteps; ++s) {
    v2f a, b;
    a.x = xrow[s * 4];
    a.y = xrow[s * 4 + 1];
    b.x = wcolk[(s * 4) * 128];
    b.y = wcolk[(s * 4 + 1) * 128];
    acc = __builtin_amdgcn_wmma_f32_16x16x4_f32(
        false, a, false, b, (short)0, acc, false, false);
  }

  // D layout: VGPR v -> row mbase + v (+8 for lanes 16-31), col = lane&15
  const int rbase = mbase + ((lane >> 4) << 3);
#pragma unroll
  for (int vv = 0; vv < 8; ++vv) {
    int r = rbase + vv;
    if (r < V_N) h[(size_t)r * HLD + col] = acc[vv];
  }
}

// ---------------- edge kernel: gaussian mixture weights + gather + scatter ----
// One wave per edge; lane = output channel m (wave32 == 32 channels).
__global__ __launch_bounds__(256) void edge_msg(
    const int* __restrict__ src, const int* __restrict__ dst,
    const float* __restrict__ ev,
    const float* __restrict__ mu, const float* __restrict__ sig,
    const float* __restrict__ h, float* __restrict__ agg) {
  int e = blockIdx.x * 8 + (threadIdx.x >> 5);
  int m = threadIdx.x & 31;
  if (e >= E_N) return;

  const float px = ev[(size_t)e * 2 + 0];
  const float py = ev[(size_t)e * 2 + 1];
  const int s = src[e];
  const float* hrow = h + (size_t)s * HLD;

  float msg = 0.0f;
#pragma unroll
  for (int k = 0; k < 3; ++k) {
    float dx = px - mu[k * 2 + 0];
    float dy = py - mu[k * 2 + 1];
    float sx = sig[k * 2 + 0];
    float sy = sig[k * 2 + 1];
    float w = __expf(-0.5f * (dx * dx / (1e-15f + sx * sx) +
                              dy * dy / (1e-15f + sy * sy)));
    msg = fmaf(w, hrow[k * 32 + m], msg);
  }
  atomicAdd(&agg[(size_t)dst[e] * 32 + m], msg);
}

// ---------------- finalize: mean aggregate + root + bias, ReLU ---------------
__global__ void finalize(const float* __restrict__ agg, const float* __restrict__ dinv,
                         const float* __restrict__ h, const float* __restrict__ bias,
                         float* __restrict__ xout) {
  int i = blockIdx.x * blockDim.x + threadIdx.x;
  if (i >= V_N * 32) return;
  int v = i >> 5, m = i & 31;
  float val = agg[i] * dinv[v] + h[(size_t)v * HLD + 96 + m] + bias[m];
  xout[i] = fmaxf(val, 0.0f);
}

// ---------------- head: fc(32->2) + log_softmax + logits[:,0] ----------------
__global__ void fc_head(const float* __restrict__ x, const float* __restrict__ w,
                        const float* __restrict__ bb,
                        float* __restrict__ ls_out, float* __restrict__ nl_out) {
  int v = blockIdx.x * blockDim.x + threadIdx.x;
  if (v >= V_N) return;
  const float* xr = x + (size_t)v * 32;
  float l0 = bb[0], l1 = bb[1];
#pragma unroll
  for (int c = 0; c < 32; ++c) {
    float xv = xr[c];
    l0 = fmaf(xv, w[c * 2 + 0], l0);
    l1 = fmaf(xv, w[c * 2 + 1], l1);
  }
  float mx = fmaxf(l0, l1);
  float lse = mx + logf(expf(l0 - mx) + expf(l1 - mx));
  ls_out[(size_t)v * 2 + 0] = l0 - lse;
  ls_out[(size_t)v * 2 + 1] = l1 - lse;
  nl_out[v] = l0;
}

// ---------------- launch ----------------

extern "C" void kernel_launch(void* const* d_in, const int* in_sizes, int n_in,
                              void* d_out, int out_size, void* d_ws, size_t ws_size,
                              hipStream_t stream) {
  (void)in_sizes; (void)n_in; (void)out_size; (void)ws_size;

  const float* data = (const float*)d_in[0];
  const int*   edges = (const int*)d_in[1];
  const int*   src = edges;
  const int*   dstp = edges + E_N;
  const float* ev = (const float*)d_in[2];

  const float* g[3]    = {(const float*)d_in[3],  (const float*)d_in[8],  (const float*)d_in[13]};
  const float* mu[3]   = {(const float*)d_in[4],  (const float*)d_in[9],  (const float*)d_in[14]};
  const float* sig[3]  = {(const float*)d_in[5],  (const float*)d_in[10], (const float*)d_in[15]};
  const float* root[3] = {(const float*)d_in[6],  (const float*)d_in[11], (const float*)d_in[16]};
  const float* bias[3] = {(const float*)d_in[7],  (const float*)d_in[12], (const float*)d_in[17]};
  const float* fcw = (const float*)d_in[18];
  const float* fcb = (const float*)d_in[19];
  float* out = (float*)d_out;

  // workspace layout (floats)
  float* ws   = (float*)d_ws;
  float* dinv = ws;                             // V
  float* x0   = dinv + V_N;                     // V*32
  float* x1   = x0 + (size_t)V_N * 32;          // V*32
  float* hbuf = x1 + (size_t)V_N * 32;          // V*128
  float* agg  = hbuf + (size_t)V_N * 128;       // V*32

  const dim3 blk(256);
  const int gV   = (V_N + 255) / 256;
  const int gE   = (E_N + 255) / 256;
  const int gVC  = (V_N * 32 + 255) / 256;
  const int gM   = (V_N + 15) / 16;       // GEMM row tiles
  const int gEw  = (E_N + 7) / 8;         // 8 edges (waves) per block

  // degree (edge structure only; shared by all layers/samples)
  zero_f32<<<gV, blk, 0, stream>>>(dinv, V_N);
  deg_count<<<gE, blk, 0, stream>>>(dstp, dinv);
  deg_invert<<<gV, blk, 0, stream>>>(dinv);

  for (int b = 0; b < 2; ++b) {
    const float* xin = data + (size_t)b * V_N * 22;
    int cin = 22, ldx = 22;
    for (int l = 0; l < 3; ++l) {
      gemm_wmma<<<gM, blk, 0, stream>>>(xin, ldx, cin, g[l], root[l], hbuf);
      zero_f32<<<gVC, blk, 0, stream>>>(agg, V_N * 32);
      edge_msg<<<gEw, blk, 0, stream>>>(src, dstp, ev, mu[l], sig[l], hbuf, agg);
      float* xout = (l & 1) ? x1 : x0;
      finalize<<<gVC, blk, 0, stream>>>(agg, dinv, hbuf, bias[l], xout);
      xin = xout; cin = 32; ldx = 32;
    }
    fc_head<<<gV, blk, 0, stream>>>(xin, fcw, fcb,
                                    out + (size_t)b * V_N * 2,
                                    out + (size_t)4 * V_N + (size_t)b * V_N);
  }
}